// ExpanderGatedGCNLayer_81149112091152
// MI455X (gfx1250) — compile-verified
//
#include <hip/hip_runtime.h>
#include <math.h>

typedef __attribute__((ext_vector_type(2))) float v2f;
typedef __attribute__((ext_vector_type(8))) float v8f;

#define DD 128
#define LDST 132          // LDS row stride in floats: 132 % 64 == 4 -> conflict-free
#define P_PART 1024       // partial-reduction rows for BN stats

// ---------------------------------------------------------------------------
// One 16x16 fp32 output tile: acc += A(16x128 in LDS) * W(128x128 row-major)
// via 32 chained V_WMMA_F32_16X16X4_F32.
// A frag (16x4): lane<16 -> a={A[l][k],A[l][k+1]}, lane>=16 -> k+2,k+3
// B frag (4x16): lane value v -> W[k + 2*half + v][colbase + lrow]
// C/D frag: element v -> row = v + 8*half, col = colbase + lrow
// ---------------------------------------------------------------------------
__device__ __forceinline__ v8f wmma_tile_f32(const float* ldsA,
                                             const float* __restrict__ W,
                                             int colbase, int lrow, int half) {
  v8f acc = {0.f, 0.f, 0.f, 0.f, 0.f, 0.f, 0.f, 0.f};
#pragma unroll
  for (int k = 0; k < DD; k += 4) {
    v2f a, b;
    const float* ap = ldsA + lrow * LDST + k + 2 * half;
    a.x = ap[0];
    a.y = ap[1];
    const int krow = k + 2 * half;
    b.x = W[(size_t)krow * DD + colbase + lrow];
    b.y = W[(size_t)(krow + 1) * DD + colbase + lrow];
    acc = __builtin_amdgcn_wmma_f32_16x16x4_f32(false, a, false, b, (short)0,
                                                acc, false, false);
  }
  return acc;
}

// ---------------------------------------------------------------------------
__global__ void zero_kernel(float4* __restrict__ p, size_t n4) {
  size_t i = (size_t)blockIdx.x * blockDim.x + threadIdx.x;
  const size_t stride = (size_t)gridDim.x * blockDim.x;
  const float4 z = {0.f, 0.f, 0.f, 0.f};
  for (; i < n4; i += stride) p[i] = z;
}

// ---------------------------------------------------------------------------
// Kernel 2: Ah,Bh,Dh,Eh = h @ {WA,WB,WD,WE} + bias.  One 16-row tile / block.
// 8 waves: wave w -> matrix (w%4), col-tiles [ (w/4)*4 .. (w/4)*4+3 ].
// ---------------------------------------------------------------------------
__global__ __launch_bounds__(256) void node_gemm_kernel(
    const float* __restrict__ h, const float* __restrict__ WA,
    const float* __restrict__ bA, const float* __restrict__ WB,
    const float* __restrict__ bB, const float* __restrict__ WD,
    const float* __restrict__ bD, const float* __restrict__ WE,
    const float* __restrict__ bE, float* __restrict__ Ah,
    float* __restrict__ Bh, float* __restrict__ Dh, float* __restrict__ Eh,
    int nrows) {
  __shared__ float lds_h[16 * LDST];
  const int row0 = blockIdx.x * 16;
  const int tid = threadIdx.x;
  // vectorized tile load: 16 rows x 32 float4 = 512 float4, 2 per thread
#pragma unroll
  for (int i = tid; i < 16 * 32; i += 256) {
    const int r = i >> 5, c4 = i & 31;
    int rr = row0 + r;
    rr = rr < nrows ? rr : nrows - 1;
    const float4 v =
        *reinterpret_cast<const float4*>(h + (size_t)rr * DD + c4 * 4);
    *reinterpret_cast<float4*>(&lds_h[r * LDST + c4 * 4]) = v;
  }
  __syncthreads();
  const int wave = tid >> 5, lane = tid & 31;
  const int lrow = lane & 15, half = lane >> 4;
  const int m = wave & 3, cg = wave >> 2;
  const float* W;
  const float* bias;
  float* out;
  switch (m) {
    case 0: W = WA; bias = bA; out = Ah; break;
    case 1: W = WB; bias = bB; out = Bh; break;
    case 2: W = WD; bias = bD; out = Dh; break;
    default: W = WE; bias = bE; out = Eh; break;
  }
  for (int ct = cg * 4; ct < cg * 4 + 4; ++ct) {
    const int colbase = ct * 16;
    v8f acc = wmma_tile_f32(lds_h, W, colbase, lrow, half);
    const float bv = bias[colbase + lrow];
#pragma unroll
    for (int v = 0; v < 8; ++v) {
      const int r = row0 + v + 8 * half;
      if (r < nrows) out[(size_t)r * DD + colbase + lrow] = acc[v] + bv;
    }
  }
}

// ---------------------------------------------------------------------------
// Kernel 3: per 16-edge tile: Ce = e@WC+bC (WMMA), then gate + scatter.
// 4 waves x 2 col-tiles.  Elementwise phase: thread == column.
// ---------------------------------------------------------------------------
__global__ __launch_bounds__(128) void edge_kernel(
    const float* __restrict__ e, const int* __restrict__ src,
    const int* __restrict__ dst, const float* __restrict__ snorm_e,
    const float* __restrict__ WC, const float* __restrict__ bC,
    const float* __restrict__ Bh, const float* __restrict__ Dh,
    const float* __restrict__ Eh, float* __restrict__ num,
    float* __restrict__ den, float* __restrict__ eij_out,
    float* __restrict__ e_psum, float* __restrict__ e_psumsq, int erows) {
  __shared__ float lds_e[16 * LDST];
  __shared__ float lds_c[16 * LDST];
  __shared__ int lds_src[16];
  __shared__ int lds_dst[16];
  __shared__ float lds_sn[16];
  const int tid = threadIdx.x;
  const int tile = blockIdx.x;
  const int row0 = tile * 16;
  // vectorized tile load: 512 float4, 4 per thread
#pragma unroll
  for (int i = tid; i < 16 * 32; i += 128) {
    const int r = i >> 5, c4 = i & 31;
    int rr = row0 + r;
    rr = rr < erows ? rr : erows - 1;
    const float4 v =
        *reinterpret_cast<const float4*>(e + (size_t)rr * DD + c4 * 4);
    *reinterpret_cast<float4*>(&lds_e[r * LDST + c4 * 4]) = v;
  }
  if (tid < 16) {
    int rr = row0 + tid;
    rr = rr < erows ? rr : erows - 1;
    lds_src[tid] = src[rr];
    lds_dst[tid] = dst[rr];
    lds_sn[tid] = snorm_e[rr];
  }
  __syncthreads();
  const int wave = tid >> 5, lane = tid & 31;
  const int lrow = lane & 15, half = lane >> 4;
#pragma unroll
  for (int t = 0; t < 2; ++t) {
    const int colbase = (wave * 2 + t) * 16;
    v8f acc = wmma_tile_f32(lds_e, WC, colbase, lrow, half);
    const float bv = bC[colbase + lrow];
#pragma unroll
    for (int v = 0; v < 8; ++v)
      lds_c[(v + 8 * half) * LDST + colbase + lrow] = acc[v] + bv;
  }
  __syncthreads();
  const int c = tid;  // column 0..127
  float ssum = 0.f, ssq = 0.f;
#pragma unroll 4
  for (int r = 0; r < 16; ++r) {
    if (row0 + r >= erows) break;
    const int s = lds_src[r], dn = lds_dst[r];
    const float val =
        lds_c[r * LDST + c] + Dh[(size_t)s * DD + c] + Eh[(size_t)dn * DD + c];
    const float sig = 1.0f / (1.0f + __expf(-val));
    atomicAdd(&num[(size_t)dn * DD + c], sig * Bh[(size_t)s * DD + c]);
    atomicAdd(&den[(size_t)dn * DD + c], sig);
    eij_out[(size_t)(row0 + r) * DD + c] = val;
    const float x = val * lds_sn[r];
    ssum += x;
    ssq += x * x;
  }
  const int prow = tile & (P_PART - 1);
  atomicAdd(&e_psum[prow * DD + c], ssum);
  atomicAdd(&e_psumsq[prow * DD + c], ssq);
}

// ---------------------------------------------------------------------------
// Kernel 4: h_pre = (Ah + num/(den+eps)) * snorm_n ; accumulate BN partials.
// ---------------------------------------------------------------------------
__global__ __launch_bounds__(128) void h_combine_kernel(
    const float* __restrict__ Ah, const float* __restrict__ num,
    const float* __restrict__ den, const float* __restrict__ snorm_n,
    float* __restrict__ hpre, float* __restrict__ h_psum,
    float* __restrict__ h_psumsq, int nrows) {
  const int c = threadIdx.x;
  const int r0 = blockIdx.x * 8;
  float ssum = 0.f, ssq = 0.f;
#pragma unroll
  for (int i = 0; i < 8; ++i) {
    const int r = r0 + i;
    if (r >= nrows) break;
    const size_t idx = (size_t)r * DD + c;
    const float v = (Ah[idx] + num[idx] / (den[idx] + 1e-6f)) * snorm_n[r];
    hpre[idx] = v;
    ssum += v;
    ssq += v * v;
  }
  const int prow = blockIdx.x & (P_PART - 1);
  atomicAdd(&h_psum[prow * DD + c], ssum);
  atomicAdd(&h_psumsq[prow * DD + c], ssq);
}

// ---------------------------------------------------------------------------
// Kernel 5: fold partials -> per-column scale/shift for both batch norms.
// ---------------------------------------------------------------------------
__global__ __launch_bounds__(256) void stats_kernel(
    const float* __restrict__ h_psum, const float* __restrict__ h_psumsq,
    const float* __restrict__ e_psum, const float* __restrict__ e_psumsq,
    const float* __restrict__ gamma_h, const float* __restrict__ beta_h,
    const float* __restrict__ gamma_e, const float* __restrict__ beta_e,
    float* __restrict__ scale_shift, float ncnt, float ecnt) {
  const int tid = threadIdx.x;
  const int c = tid & 127;
  const bool is_e = tid >= 128;
  const float* ps = is_e ? e_psum : h_psum;
  const float* pq = is_e ? e_psumsq : h_psumsq;
  const float cnt = is_e ? ecnt : ncnt;
  float s = 0.f, q = 0.f;
  for (int p = 0; p < P_PART; ++p) {
    s += ps[p * DD + c];
    q += pq[p * DD + c];
  }
  const float mu = s / cnt;
  const float var = q / cnt - mu * mu;  // biased, matches jnp.var
  const float g = is_e ? gamma_e[c] : gamma_h[c];
  const float b = is_e ? beta_e[c] : beta_h[c];
  const float sc = g * rsqrtf(var + 1e-5f);
  const float sh = b - mu * sc;
  scale_shift[(is_e ? 2 : 0) * DD + c] = sc;
  scale_shift[(is_e ? 3 : 1) * DD + c] = sh;
}

// ---------------------------------------------------------------------------
// Kernel 6: out = residual + relu(pre * scale + shift), in place over d_out.
// Vectorized float4: nh is a multiple of 128, so the h/e split is aligned.
// ---------------------------------------------------------------------------
__global__ __launch_bounds__(256) void apply_kernel(
    const float4* __restrict__ h4, const float4* __restrict__ e4,
    const float4* __restrict__ ss4, float4* __restrict__ out4, size_t nh4,
    size_t ne4) {
  size_t i = (size_t)blockIdx.x * 256 + threadIdx.x;
  const size_t total4 = nh4 + ne4;
  const size_t stride = (size_t)gridDim.x * 256;
  for (; i < total4; i += stride) {
    const int c4 = (int)(i & 31);  // float4-column within the 128-wide row
    const float4 pre = out4[i];
    float4 sc, sh, res;
    if (i < nh4) {
      sc = ss4[c4];
      sh = ss4[32 + c4];
      res = h4[i];
    } else {
      sc = ss4[64 + c4];
      sh = ss4[96 + c4];
      res = e4[i - nh4];
    }
    float4 o;
    o.x = res.x + fmaxf(pre.x * sc.x + sh.x, 0.f);
    o.y = res.y + fmaxf(pre.y * sc.y + sh.y, 0.f);
    o.z = res.z + fmaxf(pre.z * sc.z + sh.z, 0.f);
    o.w = res.w + fmaxf(pre.w * sc.w + sh.w, 0.f);
    out4[i] = o;
  }
}

// ---------------------------------------------------------------------------
extern "C" void kernel_launch(void* const* d_in, const int* in_sizes, int n_in,
                              void* d_out, int out_size, void* d_ws,
                              size_t ws_size, hipStream_t stream) {
  const float* h = (const float*)d_in[0];
  const float* e = (const float*)d_in[1];
  const int* src = (const int*)d_in[2];
  const int* dst = (const int*)d_in[3];
  const float* snorm_n = (const float*)d_in[4];
  const float* snorm_e = (const float*)d_in[5];
  const float* WA = (const float*)d_in[6];
  const float* bA = (const float*)d_in[7];
  const float* WB = (const float*)d_in[8];
  const float* bB = (const float*)d_in[9];
  const float* WC = (const float*)d_in[10];
  const float* bC = (const float*)d_in[11];
  const float* WD = (const float*)d_in[12];
  const float* bD = (const float*)d_in[13];
  const float* WE = (const float*)d_in[14];
  const float* bE = (const float*)d_in[15];
  const float* gamma_h = (const float*)d_in[16];
  const float* beta_h = (const float*)d_in[17];
  const float* gamma_e = (const float*)d_in[18];
  const float* beta_e = (const float*)d_in[19];

  const int N = in_sizes[0] / DD;
  const int E = in_sizes[1] / DD;
  const size_t nd = (size_t)N * DD;
  const size_t ed = (size_t)E * DD;

  // workspace layout (floats)
  float* ws = (float*)d_ws;
  float* Ah = ws;
  float* Bh = Ah + nd;
  float* Dh = Bh + nd;
  float* Eh = Dh + nd;
  float* num = Eh + nd;
  float* den = num + nd;
  float* h_psum = den + nd;
  float* h_psumsq = h_psum + (size_t)P_PART * DD;
  float* e_psum = h_psumsq + (size_t)P_PART * DD;
  float* e_psumsq = e_psum + (size_t)P_PART * DD;
  float* scale_shift = e_psumsq + (size_t)P_PART * DD;

  float* hpre = (float*)d_out;      // h region of output, used as scratch
  float* eij = (float*)d_out + nd;  // e region of output, used as scratch

  // 1) zero num, den, partials, scale_shift (contiguous from num)
  const size_t zn4 = (2 * nd + 4 * (size_t)P_PART * DD + 4 * DD) / 4;
  zero_kernel<<<4096, 256, 0, stream>>>((float4*)num, zn4);

  // 2) node GEMMs (WMMA f32)
  node_gemm_kernel<<<(N + 15) / 16, 256, 0, stream>>>(
      h, WA, bA, WB, bB, WD, bD, WE, bE, Ah, Bh, Dh, Eh, N);

  // 3) edge GEMM + gate + scatter (WMMA f32)
  edge_kernel<<<(E + 15) / 16, 128, 0, stream>>>(e, src, dst, snorm_e, WC, bC,
                                                 Bh, Dh, Eh, num, den, eij,
                                                 e_psum, e_psumsq, E);

  // 4) node combine
  h_combine_kernel<<<(N + 7) / 8, 128, 0, stream>>>(Ah, num, den, snorm_n,
                                                    hpre, h_psum, h_psumsq, N);

  // 5) batchnorm stats
  stats_kernel<<<1, 256, 0, stream>>>(h_psum, h_psumsq, e_psum, e_psumsq,
                                      gamma_h, beta_h, gamma_e, beta_e,
                                      scale_shift, (float)N, (float)E);

  // 6) bn-apply + relu + residual, in place on d_out (float4)
  apply_kernel<<<26624, 256, 0, stream>>>((const float4*)h, (const float4*)e,
                                          (const float4*)scale_shift,
                                          (float4*)d_out, nd / 4, ed / 4);
}